// FeatureEncoder_73023033967269
// MI455X (gfx1250) — compile-verified
//
#include <hip/hip_runtime.h>
#include <math.h>

// FeatureEncoder: per-row masked matvec (1x64 @ 64x64) + bias + tanh-GELU.
// HBM-bound (AI ~0.5 FLOP/B). Key optimizations:
//  * skip W rows >= lengths[n]  (halves average W traffic: 819MB -> ~455MB)
//  * stream W with non-temporal loads (no reuse; keep 192MB L2 clean)
//  * compute via V_WMMA_F32_16X16X4_F32, one row per wave32:
//      A = W^T tile (16x4), B = x chunk broadcast to all 16 columns,
//      C accumulates partial y; all D columns identical (waste is free here).
//  * branch-free epilogue: register-select + ds_bpermute half swap, no LDS,
//    no EXEC-mask toggling (previous version spent ~160 salu instr here).

typedef __attribute__((ext_vector_type(2))) float v2f;
typedef __attribute__((ext_vector_type(8))) float v8f;

#define MAX_IN 64
#define DM 64
#define WAVES_PER_BLOCK 8

__device__ __forceinline__ float gelu_tanh(float x) {
    const float k0 = 0.7978845608028654f;  // sqrt(2/pi)
    const float k1 = 0.044715f;
    float x3 = x * x * x;
    float t = tanhf(k0 * (x + k1 * x3));
    return 0.5f * x * (1.0f + t);
}

__global__ __launch_bounds__(WAVES_PER_BLOCK * 32)
void featenc_wmma_kernel(const float* __restrict__ seg,
                         const float* __restrict__ W,
                         const float* __restrict__ bias,
                         const int* __restrict__ lengths,
                         float* __restrict__ out, int n)
{
    const int lane = threadIdx.x & 31;
    const int wv   = threadIdx.x >> 5;
    const int row  = blockIdx.x * WAVES_PER_BLOCK + wv;
    if (row >= n) return;                 // wave-uniform

    const int len  = lengths[row];        // wave-uniform
    const int half = lane >> 4;           // which 16-lane half
    const int m    = lane & 15;

    // Lane L caches x[2L], x[2L+1] via one b64 load, masked beyond len.
    const float* segp = seg + (size_t)row * MAX_IN;
    v2f xv = __builtin_nontemporal_load((const v2f*)segp + lane);
    float x0 = (2 * lane     < len) ? xv.x : 0.0f;
    float x1 = (2 * lane + 1 < len) ? xv.y : 0.0f;

    v8f acc[4] = {};                      // 4 N-blocks of 16 outputs each

    const float* Wp = W + (size_t)row * (MAX_IN * DM);
    const int kblocks = (len + 3) >> 2;   // skip masked K entirely (bandwidth win)

    for (int ko = 0; ko < kblocks; ++ko) {
        // B (4x16) = x[4ko..4ko+3] broadcast across all 16 columns.
        // ISA B layout: VGPR0 = rows K0|K2, VGPR1 = rows K1|K3 -> lane needs
        // x[4ko+2*half], x[4ko+2*half+1], held as the float2 of lane 2ko+half.
        int src = 2 * ko + half;
        v2f bmat;
        bmat.x = __shfl(x0, src, 32);
        bmat.y = __shfl(x1, src, 32);

        // A (16x4) = W[4ko+k, 16mo+m] (transposed tile): lane m reads column m.
        // VGPR0 holds K = 2*half, VGPR1 holds K = 2*half+1 -> rows r and r+1.
        const float* wr = Wp + (size_t)(4 * ko + 2 * half) * DM + m;
        #pragma unroll
        for (int mo = 0; mo < 4; ++mo) {
            v2f amat;
            amat.x = __builtin_nontemporal_load(wr + 16 * mo);        // row r
            amat.y = __builtin_nontemporal_load(wr + DM + 16 * mo);   // row r+1
            // D[m,n] = sum_k W[4ko+k,16mo+m] * x[4ko+k]  (same for every n)
            acc[mo] = __builtin_amdgcn_wmma_f32_16x16x4_f32(
                false, amat, false, bmat, (short)0, acc[mo], false, false);
        }
    }

    // ---- Branch-free epilogue ----
    // C/D layout: acc[mo][j] = y[16*mo + 8*half + j] (identical across N lanes).
    // Step 1: per mo, register-select j = lane&7 (cndmask chain, masks shared).
    // Step 2: ds_bpermute from lane^16 supplies the other half's value; select
    //         so lane holds val[mo] = y[16*mo + (lane&15)].
    const int jsel    = lane & 7;
    const int ownHalf = (lane >> 3) & 1;  // which 8-row group lane's m falls in
    float val[4];
    #pragma unroll
    for (int mo = 0; mo < 4; ++mo) {
        float t = acc[mo][0];
        t = (jsel == 1) ? acc[mo][1] : t;
        t = (jsel == 2) ? acc[mo][2] : t;
        t = (jsel == 3) ? acc[mo][3] : t;
        t = (jsel == 4) ? acc[mo][4] : t;
        t = (jsel == 5) ? acc[mo][5] : t;
        t = (jsel == 6) ? acc[mo][6] : t;
        t = (jsel == 7) ? acc[mo][7] : t;
        float other = __shfl(t, lane ^ 16, 32);   // other half, same j
        val[mo] = (ownHalf == half) ? t : other;
    }

    // Step 3: lanes 0-15 store d in [0,32), lanes 16-31 store d in [32,64).
    const int d0 = 32 * half + m;              // = 16*(2*half) + m
    float vA = half ? val[2] : val[0];         // y[d0]
    float vB = half ? val[3] : val[1];         // y[d0+16]

    const float* bp = bias + (size_t)row * DM;
    float y0 = vA + __builtin_nontemporal_load(&bp[d0]);
    float y1 = vB + __builtin_nontemporal_load(&bp[d0 + 16]);

    float* op = out + (size_t)row * DM;
    __builtin_nontemporal_store(gelu_tanh(y0), &op[d0]);
    __builtin_nontemporal_store(gelu_tanh(y1), &op[d0 + 16]);
}

extern "C" void kernel_launch(void* const* d_in, const int* in_sizes, int n_in,
                              void* d_out, int out_size, void* d_ws, size_t ws_size,
                              hipStream_t stream) {
    const float* seg = (const float*)d_in[0];   // [N, 64] f32
    const float* W   = (const float*)d_in[1];   // [N, 64, 64] f32
    const float* b   = (const float*)d_in[2];   // [N, 64] f32
    const int*   len = (const int*)d_in[3];     // [N] i32
    float* out = (float*)d_out;                 // [N, 64] f32

    const int n = in_sizes[3];                  // N from lengths
    const int blocks = (n + WAVES_PER_BLOCK - 1) / WAVES_PER_BLOCK;
    featenc_wmma_kernel<<<blocks, WAVES_PER_BLOCK * 32, 0, stream>>>(
        seg, W, b, len, out, n);
}